// Equilibrium_Propagation_Network_62139586839109
// MI455X (gfx1250) — compile-verified
//
#include <hip/hip_runtime.h>

typedef float v2f __attribute__((ext_vector_type(2)));
typedef float v8f __attribute__((ext_vector_type(8)));

#define HIDDEN   128
#define OUTSZ    1024
#define INSZ     262144
#define NB1      512          // phase-1 blocks
#define RPB      512          // rows of W0 per block (NB1*RPB == INSZ)
#define LRC      0.01f
#define BETA1    0.9f
#define BETA2    0.999f
#define EPSA     1e-8f

__device__ __forceinline__ float rho(float s) { return fminf(fmaxf(s, 0.f), 1.f); }
// jax clip gradient: 1 inside, 0 outside, 0.5 at the tie points (lax balanced-eq)
__device__ __forceinline__ float rhop(float s) {
    if (s > 0.f && s < 1.f) return 1.f;
    if (s == 0.f || s == 1.f) return 0.5f;
    return 0.f;
}

// ---------------------------------------------------------------------------
// Phase 1: a = rho(x)^T @ W0  via V_WMMA_F32_16X16X4_F32 (exact fp32 path).
// Each block: 512 rows of W0. 8 waves, wave w owns 16-column tile n0 = 16*w.
// Per WMMA step: A(16x4) = rho(x[row..row+3]) broadcast across M,
//                B(4x16)  = W0[row..row+3][n0..n0+15].
// All 16 rows of D accumulate identical column sums -> read acc[0], lanes 0-15.
// ---------------------------------------------------------------------------
__global__ __launch_bounds__(256)
void ep_phase1(const float* __restrict__ x,
               const float* __restrict__ W0,
               float* __restrict__ partial) {
    __shared__ float rx[RPB];
    const int tid = threadIdx.x;
    const int blk = blockIdx.x;
    const int rowBase = blk * RPB;

    rx[tid]       = rho(x[rowBase + tid]);
    rx[tid + 256] = rho(x[rowBase + tid + 256]);
    __syncthreads();

    const int wave = tid >> 5;
    const int lane = tid & 31;
    const int half = lane >> 4;     // 0: K=0,1  1: K=2,3
    const int ln   = lane & 15;
    const int n0   = wave * 16;

    // per-lane W0 pointer for B: row (rowBase + half*2), column n0+ln
    const float* wp = W0 + (size_t)(rowBase + half * 2) * HIDDEN + n0 + ln;

    v8f acc = {0.f, 0.f, 0.f, 0.f, 0.f, 0.f, 0.f, 0.f};
    #pragma unroll 4
    for (int r = 0; r < RPB; r += 4) {
        v2f a, b;
        a.x = rx[r + half * 2];         // K = 0 (or 2)
        a.y = rx[r + half * 2 + 1];     // K = 1 (or 3)
        b.x = wp[0];                    // W0[row + half*2    ][n0+ln]
        b.y = wp[HIDDEN];               // W0[row + half*2 + 1][n0+ln]
        acc = __builtin_amdgcn_wmma_f32_16x16x4_f32(
                  false, a, false, b, (short)0, acc, false, false);
        wp += 4 * HIDDEN;
    }
    // D row 0, columns n0..n0+15 live in lanes 0..15 of acc[0]
    if (lane < 16)
        partial[(size_t)blk * HIDDEN + n0 + lane] = acc[0];
}

// ---------------------------------------------------------------------------
// Phase 2: deterministic reduce of partials + 20 Adam iterations on (h, o).
// Single block of 1024 threads; state in registers, rh/ro in LDS.
// ---------------------------------------------------------------------------
__global__ __launch_bounds__(1024)
void ep_phase2(const float* __restrict__ partial,
               const float* __restrict__ b1v,
               const float* __restrict__ b2v,
               const float* __restrict__ W1,
               const int* __restrict__ n_iter_p,
               float* __restrict__ out) {
    __shared__ float rh[HIDDEN];
    __shared__ float ro[OUTSZ];
    const int t = threadIdx.x;
    const int niter = *n_iter_p;

    float a_t = 0.f, b1_t = 0.f;
    float h = 0.f, mh = 0.f, vh = 0.f;
    if (t < HIDDEN) {
        for (int b = 0; b < NB1; ++b)          // fixed-order reduction
            a_t += partial[b * HIDDEN + t];
        b1_t = b1v[t];
    }
    const float b2_t = b2v[t];
    float o = 0.f, mo = 0.f, vo = 0.f;
    float p1 = 1.f, p2 = 1.f;                  // beta^t running powers

    for (int it = 0; it < niter; ++it) {
        p1 *= BETA1; p2 *= BETA2;
        if (t < HIDDEN) rh[t] = rho(h);
        ro[t] = rho(o);
        __syncthreads();

        // go: v_k = sum_j rh[j] * W1[j][k]  (coalesced across threads)
        float v = 0.f;
        #pragma unroll 8
        for (int j = 0; j < HIDDEN; ++j)
            v = fmaf(rh[j], W1[(size_t)j * OUTSZ + t], v);
        const float go = rhop(o) * (ro[t] - b2_t - v);

        // gh: u_j = sum_k W1[j][k] * ro[k]  (float4 row stream from L2)
        float gh = 0.f;
        if (t < HIDDEN) {
            float u = 0.f;
            const float4* w = (const float4*)(W1 + (size_t)t * OUTSZ);
            #pragma unroll 4
            for (int c = 0; c < OUTSZ / 4; ++c) {
                const float4 f = w[c];
                u = fmaf(f.x, ro[4 * c + 0], u);
                u = fmaf(f.y, ro[4 * c + 1], u);
                u = fmaf(f.z, ro[4 * c + 2], u);
                u = fmaf(f.w, ro[4 * c + 3], u);
            }
            gh = rhop(h) * (rh[t] - b1_t - a_t - u);
        }

        // Adam (torch defaults), bias-corrected
        const float ic1 = 1.f / (1.f - p1);
        const float ic2 = 1.f / (1.f - p2);
        if (t < HIDDEN) {
            mh = BETA1 * mh + (1.f - BETA1) * gh;
            vh = BETA2 * vh + (1.f - BETA2) * gh * gh;
            h  = h - LRC * (mh * ic1) / (sqrtf(vh * ic2) + EPSA);
        }
        mo = BETA1 * mo + (1.f - BETA1) * go;
        vo = BETA2 * vo + (1.f - BETA2) * go * go;
        o  = o - LRC * (mo * ic1) / (sqrtf(vo * ic2) + EPSA);
        __syncthreads();                       // protect rh/ro for next iter
    }
    out[t] = o;
}

// ---------------------------------------------------------------------------
// inputs: x, b0, b1, b2, W0, W1, n_iterations   (b0 never enters gradients)
// ---------------------------------------------------------------------------
extern "C" void kernel_launch(void* const* d_in, const int* in_sizes, int n_in,
                              void* d_out, int out_size, void* d_ws, size_t ws_size,
                              hipStream_t stream) {
    const float* x   = (const float*)d_in[0];
    const float* b1v = (const float*)d_in[2];
    const float* b2v = (const float*)d_in[3];
    const float* W0  = (const float*)d_in[4];
    const float* W1  = (const float*)d_in[5];
    const int*   nit = (const int*)d_in[6];

    float* partial = (float*)d_ws;            // NB1*HIDDEN floats = 256 KB

    ep_phase1<<<NB1, 256, 0, stream>>>(x, W0, partial);
    ep_phase2<<<1, 1024, 0, stream>>>(partial, b1v, b2v, W1, nit, (float*)d_out);
}